// ConvLayer_49864570307077
// MI455X (gfx1250) — compile-verified
//
#include <hip/hip_runtime.h>

typedef __attribute__((ext_vector_type(2))) float v2f;
typedef __attribute__((ext_vector_type(8))) float v8f;

#define FDIM 64
#define CDIM 128
#define TILE_M 16
#define LDS_STRIDE 68           // 64 + 4 floats pad -> conflict-free ds_load_b64
#define TILES_PER_BLOCK 32      // 512 rows per block
#define NBLOCKS 2048            // 2048 * 512 = 1,048,576 rows
#define TILE_BYTES (TILE_M * FDIM * 4)   // 4 KB of x per tile

__global__ void zero_out_kernel(float* __restrict__ out, int n) {
    int i = blockIdx.x * blockDim.x + threadIdx.x;
    if (i < n) out[i] = 0.0f;
}

// Flush per-wave accumulator (16 rows x 16 cols, D-layout) into out[cur_seg].
// acc[i]: lane<16 holds row i, lane>=16 holds row i+8, column = colbase+(lane&15).
__device__ __forceinline__ void flush_acc(float* __restrict__ out, int cur_seg,
                                          const v8f& acc, int col, int lane) {
    if (cur_seg < 0) return;
    float p = acc[0] + acc[1] + acc[2] + acc[3] + acc[4] + acc[5] + acc[6] + acc[7];
    p += __shfl_xor(p, 16, 32);                 // fold the two 8-row halves
    if (lane < 16) atomicAdd(&out[(size_t)cur_seg * CDIM + col], p);
}

// Async copy of one 16-byte chunk from global memory straight into LDS
// (gfx1250 GLOBAL_LOAD_ASYNC_TO_LDS_B128, tracked by ASYNCcnt).
#define ASYNC_TILE_TO_LDS(LDSOFF, GADDR)                                      \
    asm volatile("global_load_async_to_lds_b128 %0, %1, off"                  \
                 :: "v"(LDSOFF), "v"(GADDR) : "memory")

__global__ __launch_bounds__(256)
void fused_gemm_sigmoid_segsum(const float* __restrict__ x,
                               const float* __restrict__ w,
                               const int*   __restrict__ seg,
                               float*       __restrict__ out) {
    __shared__ __align__(16) float buf[2][TILE_M * LDS_STRIDE];

    const int tid     = threadIdx.x;
    const int lane    = tid & 31;
    const int wave    = tid >> 5;        // 0..7 -> 16-column slice of C
    const int laneHi  = lane >> 4;       // 0/1 lane half
    const int lane15  = lane & 15;
    const int colbase = wave * 16;
    const int col     = colbase + lane15;
    const int block_row0 = blockIdx.x * (TILES_PER_BLOCK * TILE_M);

    // ---- Preload B fragments: w[64,128] slice for this wave's 16 columns ----
    // 32-bit B layout (4x16 K x N): lanes0-15 N=0..15, VGPR0 K=k0, VGPR1 K=k0+1,
    // lane half selects K pair (k0 = 4k + 2*laneHi).
    v2f b[16];
#pragma unroll
    for (int k = 0; k < 16; ++k) {
        const int kk = 4 * k + 2 * laneHi;
        v2f t;
        t.x = w[(size_t)kk * CDIM + col];
        t.y = w[(size_t)(kk + 1) * CDIM + col];
        b[k] = t;
    }

    // ---- Double-buffered async staging of x tiles into LDS ----
    const int r  = tid >> 4;            // 0..15 row within tile
    const int s4 = tid & 15;            // 0..15 float4 segment within row
    const unsigned long long gbase =
        (unsigned long long)(uintptr_t)(x + (size_t)(block_row0 + r) * FDIM + s4 * 4);
    const unsigned lds0 = (unsigned)(uintptr_t)&buf[0][r * LDS_STRIDE + s4 * 4];
    const unsigned lds1 = (unsigned)(uintptr_t)&buf[1][r * LDS_STRIDE + s4 * 4];

    ASYNC_TILE_TO_LDS(lds0, gbase);                                   // tile 0
    ASYNC_TILE_TO_LDS(lds1, gbase + (unsigned long long)TILE_BYTES);  // tile 1

    v8f acc = {0.f, 0.f, 0.f, 0.f, 0.f, 0.f, 0.f, 0.f};
    int cur_seg = -1;

    for (int t = 0; t < TILES_PER_BLOCK; ++t) {
        const int row0 = block_row0 + t * TILE_M;

        // Tile t landed when at most one newer async load is outstanding
        // (async loads from a wave complete in order).
        if (t < TILES_PER_BLOCK - 1)
            asm volatile("s_wait_asynccnt 0x1" ::: "memory");
        else
            asm volatile("s_wait_asynccnt 0x0" ::: "memory");
        __syncthreads();   // all waves' chunks of tile t are in LDS

        // ---- 16x16 output tile, K=64: two independent wmma chains of 8 ----
        // A layout (16x4 M x K): lane holds row M=lane&15, K pair = 4k + 2*laneHi.
        const float* abase = &buf[t & 1][lane15 * LDS_STRIDE + 2 * laneHi];
        v8f d0 = {0.f, 0.f, 0.f, 0.f, 0.f, 0.f, 0.f, 0.f};
        v8f d1 = {0.f, 0.f, 0.f, 0.f, 0.f, 0.f, 0.f, 0.f};
#pragma unroll
        for (int k = 0; k < 8; ++k) {
            v2f a0 = *(const v2f*)(abase + 8 * k);
            v2f a1 = *(const v2f*)(abase + 8 * k + 4);
            d0 = __builtin_amdgcn_wmma_f32_16x16x4_f32(
                     false, a0, false, b[2 * k], (short)0, d0, false, false);
            d1 = __builtin_amdgcn_wmma_f32_16x16x4_f32(
                     false, a1, false, b[2 * k + 1], (short)0, d1, false, false);
        }

        // ---- sigmoid (v_exp_f32 + v_rcp_f32) on merged accumulators ----
        v8f s;
#pragma unroll
        for (int i = 0; i < 8; ++i)
            s[i] = __builtin_amdgcn_rcpf(1.0f + __expf(-(d0[i] + d1[i])));

        // ---- segmented accumulation (ids sorted; row0 uniform -> s_loads) ----
        const int s0  = seg[row0];
        const int s15 = seg[row0 + 15];
        if (s0 == s15) {                       // whole tile in one bag (common)
            if (s0 != cur_seg) {
                flush_acc(out, cur_seg, acc, col, lane);
                cur_seg = s0;
                acc = s;
            } else {
#pragma unroll
                for (int i = 0; i < 8; ++i) acc[i] += s[i];
            }
        } else {                               // bag boundary inside tile (rare)
            flush_acc(out, cur_seg, acc, col, lane);
            cur_seg = -1;
#pragma unroll
            for (int i = 0; i < 8; ++i) acc[i] = 0.f;
#pragma unroll
            for (int i = 0; i < 8; ++i) {
                const int row = row0 + i + 8 * laneHi;   // D-layout row of acc[i]
                const int sg  = seg[row];
                atomicAdd(&out[(size_t)sg * CDIM + col], s[i]);
            }
        }

        __syncthreads();   // all waves done reading buf[t&1]
        if (t + 2 < TILES_PER_BLOCK) {
            const unsigned dst = (t & 1) ? lds1 : lds0;   // buf[(t+2)&1] == buf[t&1]
            ASYNC_TILE_TO_LDS(dst, gbase + (unsigned long long)(t + 2) * TILE_BYTES);
        }
    }
    flush_acc(out, cur_seg, acc, col, lane);
}

extern "C" void kernel_launch(void* const* d_in, const int* in_sizes, int n_in,
                              void* d_out, int out_size, void* d_ws, size_t ws_size,
                              hipStream_t stream) {
    const float* x   = (const float*)d_in[0];   // [N, 64] f32
    const float* w   = (const float*)d_in[1];   // [64, 128] f32
    const int*   seg = (const int*)d_in[2];     // [N] i32, sorted
    // d_in[3] = num_segments (fixed 4096, unused)
    float* out = (float*)d_out;                 // [4096, 128] f32

    zero_out_kernel<<<(out_size + 255) / 256, 256, 0, stream>>>(out, out_size);
    fused_gemm_sigmoid_segsum<<<NBLOCKS, 256, 0, stream>>>(x, w, seg, out);
}